// AlphaCompositor_70635032150717
// MI455X (gfx1250) — compile-verified
//
#include <hip/hip_runtime.h>
#include <stdint.h>

// Problem dims (fixed by the reference).
constexpr int N_  = 8;
constexpr int K_  = 16;
constexpr int H_  = 256;
constexpr int W_  = 256;
constexpr int C_  = 4;
constexpr int HW  = H_ * W_;        // 65536
constexpr int NHW = N_ * HW;        // 524288

constexpr int PXT  = 4;             // pixels per thread (b128 loads)
constexpr int TPB  = 128;           // 4 wave32 per block
constexpr int WAVE = 32;

// Native clang vectors (usable with nontemporal builtins).
typedef float v4f __attribute__((ext_vector_type(4)));
typedef int   v4i __attribute__((ext_vector_type(4)));

// LDS layout per wave: frag[K][32 lanes][16B] then alpha[K][32 lanes][16B]
constexpr int SLICE_BYTES = WAVE * 16;                  // 512 B per k-slice per matrix
constexpr int WAVE_REGION = 2 * K_ * SLICE_BYTES;       // 16 KB per wave
constexpr int LDS_BYTES   = (TPB / WAVE) * WAVE_REGION; // 64 KB per block -> 5 blocks/WGP

// ---------------------------------------------------------------------------
// Pre-pass: transpose ptclds (C,P) -> (P, float4) so each gather is one 16B
// L2 hit instead of four scattered 4B loads.
// ---------------------------------------------------------------------------
__global__ void AlphaComp_transpose_kernel(const float* __restrict__ src,
                                           v4f* __restrict__ dst, int P) {
    int p = blockIdx.x * blockDim.x + threadIdx.x;
    if (p < P) {
        v4f v;
        v.x = src[p];
        v.y = src[P + p];
        v.z = src[2 * P + p];
        v.w = src[3 * P + p];
        dst[p] = v;
    }
}

// ---------------------------------------------------------------------------
// Composite step for one pixel, one k.
// ---------------------------------------------------------------------------
template <bool USE_TR>
__device__ __forceinline__ void comp_step(int fj, float aj, int P,
                                          const float* __restrict__ ptclds,
                                          const v4f* __restrict__ ptcldsT,
                                          float& T, v4f& acc) {
    const bool  v  = (fj >= 0);
    const float av = v ? aj : 0.0f;
    const float wk = av * T;
    T *= (1.0f - av);
    if (v) {
        v4f ft;
        if (USE_TR) {
            ft = ptcldsT[fj];
        } else {
            ft.x = ptclds[fj];
            ft.y = ptclds[P + fj];
            ft.z = ptclds[2 * P + fj];
            ft.w = ptclds[3 * P + fj];
        }
        acc.x = fmaf(wk, ft.x, acc.x);
        acc.y = fmaf(wk, ft.y, acc.y);
        acc.z = fmaf(wk, ft.z, acc.z);
        acc.w = fmaf(wk, ft.w, acc.w);
    }
}

// ---------------------------------------------------------------------------
// Main kernel: async-DMA the 16 strided k-slices (frag + alpha) into LDS with
// NT hint (streamed once; keep the gather table resident in L2), composite
// from LDS, gather features from the L2-resident table, NT b128 stores.
// ---------------------------------------------------------------------------
template <bool USE_TR>
__global__ __launch_bounds__(TPB) void AlphaCompositor_70635032150717_kernel(
    const int*   __restrict__ fragments,   // (N,K,H,W)
    const float* __restrict__ alphas,      // (N,K,H,W)
    const float* __restrict__ ptclds,      // (C,P)
    const float* __restrict__ bgc,         // (3,)
    const v4f*   __restrict__ ptcldsT,     // (P,4) in ws (if USE_TR)
    int P,
    float* __restrict__ out)               // (N,C,H,W)
{
    __shared__ __align__(16) unsigned char smem[LDS_BYTES];

    const int tid = blockIdx.x * TPB + threadIdx.x;
    const int px  = tid * PXT;                 // first of 4 consecutive w-pixels
    if (px >= NHW) return;
    const int n   = px / HW;
    const int rem = px - n * HW;               // multiple of 4 -> 16B aligned

    const int lane = threadIdx.x & (WAVE - 1);
    const int wave = threadIdx.x >> 5;

    // HW LDS byte address (low 32 bits of the flat address of a __shared__
    // object are the LDS offset the async engine needs).
    const uint32_t waveBase = (uint32_t)(uintptr_t)(&smem[wave * WAVE_REGION]);
    const uint32_t ldsF = waveBase + (uint32_t)(lane * 16);
    const uint32_t ldsA = ldsF + (uint32_t)(K_ * SLICE_BYTES);

    const int*   fbase = fragments + (size_t)n * K_ * HW;
    const float* abase = alphas    + (size_t)n * K_ * HW;

    // NOTE: the async path adds the immediate offset to BOTH the global and
    // the LDS address, so k is baked into the VGPR addresses (imm = 0).
#pragma unroll
    for (int k = 0; k < K_; ++k) {
        const uint32_t go = (uint32_t)(rem * 4) + (uint32_t)(k * HW * 4);
        asm volatile("global_load_async_to_lds_b128 %0, %1, %2 th:TH_LOAD_NT"
                     :: "v"(ldsF + (uint32_t)(k * SLICE_BYTES)), "v"(go), "s"(fbase)
                     : "memory");
        asm volatile("global_load_async_to_lds_b128 %0, %1, %2 th:TH_LOAD_NT"
                     :: "v"(ldsA + (uint32_t)(k * SLICE_BYTES)), "v"(go), "s"(abase)
                     : "memory");
    }

    // Uniform scalar work while the DMA is in flight.
    const float bg0 = bgc[0], bg1 = bgc[1], bg2 = bgc[2];

    asm volatile("s_wait_asynccnt 0x0" ::: "memory");

    const unsigned char* fr = smem + wave * WAVE_REGION + lane * 16;
    const unsigned char* ar = fr + K_ * SLICE_BYTES;

    float T0 = 1.0f, T1 = 1.0f, T2 = 1.0f, T3 = 1.0f;
    v4f acc0 = {0.f, 0.f, 0.f, 0.f};
    v4f acc1 = {0.f, 0.f, 0.f, 0.f};
    v4f acc2 = {0.f, 0.f, 0.f, 0.f};
    v4f acc3 = {0.f, 0.f, 0.f, 0.f};
    v4i f0   = {0, 0, 0, 0};

#pragma unroll
    for (int k = 0; k < K_; ++k) {
        const v4i f = *(const v4i*)(fr + k * SLICE_BYTES);
        const v4f a = *(const v4f*)(ar + k * SLICE_BYTES);
        if (k == 0) f0 = f;
        comp_step<USE_TR>(f.x, a.x, P, ptclds, ptcldsT, T0, acc0);
        comp_step<USE_TR>(f.y, a.y, P, ptclds, ptcldsT, T1, acc1);
        comp_step<USE_TR>(f.z, a.z, P, ptclds, ptcldsT, T2, acc2);
        comp_step<USE_TR>(f.w, a.w, P, ptclds, ptcldsT, T3, acc3);
    }

    // Background fill where no nearest point exists (k=0 fragment < 0).
    const v4f bgv = {bg0, bg1, bg2, 1.0f};
    if (f0.x < 0) acc0 = bgv;
    if (f0.y < 0) acc1 = bgv;
    if (f0.z < 0) acc2 = bgv;
    if (f0.w < 0) acc3 = bgv;

    // Transpose 4 pixels x 4 channels -> one b128 NT store per channel.
    float* obase = out + (size_t)n * C_ * HW + rem;
    const v4f r0 = {acc0.x, acc1.x, acc2.x, acc3.x};
    const v4f r1 = {acc0.y, acc1.y, acc2.y, acc3.y};
    const v4f r2 = {acc0.z, acc1.z, acc2.z, acc3.z};
    const v4f r3 = {acc0.w, acc1.w, acc2.w, acc3.w};
    __builtin_nontemporal_store(r0, (v4f*)(obase));
    __builtin_nontemporal_store(r1, (v4f*)(obase + HW));
    __builtin_nontemporal_store(r2, (v4f*)(obase + 2 * HW));
    __builtin_nontemporal_store(r3, (v4f*)(obase + 3 * HW));
}

// ---------------------------------------------------------------------------
extern "C" void kernel_launch(void* const* d_in, const int* in_sizes, int n_in,
                              void* d_out, int out_size, void* d_ws, size_t ws_size,
                              hipStream_t stream) {
    const int*   fragments = (const int*)d_in[0];
    const float* alphas    = (const float*)d_in[1];
    const float* ptclds    = (const float*)d_in[2];
    const float* bgc       = (const float*)d_in[3];
    float*       out       = (float*)d_out;

    const int P = in_sizes[2] / C_;   // 100000

    v4f* ptcldsT = (v4f*)d_ws;
    const bool use_tr = (ws_size >= (size_t)P * sizeof(v4f));

    const int threads = NHW / PXT;    // 131072
    if (use_tr) {
        AlphaComp_transpose_kernel<<<(P + 255) / 256, 256, 0, stream>>>(
            ptclds, ptcldsT, P);
        AlphaCompositor_70635032150717_kernel<true><<<threads / TPB, TPB, 0, stream>>>(
            fragments, alphas, ptclds, bgc, ptcldsT, P, out);
    } else {
        AlphaCompositor_70635032150717_kernel<false><<<threads / TPB, TPB, 0, stream>>>(
            fragments, alphas, ptclds, bgc, ptcldsT, P, out);
    }
}